// E2VAggLayer_25589415150165
// MI455X (gfx1250) — compile-verified
//
#include <hip/hip_runtime.h>

typedef float v2f __attribute__((ext_vector_type(2)));
typedef float v8f __attribute__((ext_vector_type(8)));

#define V_NUM   100000
#define E_NUM   50000
#define N_PAIRS 600000
#define DIM     128
#define INV_TAU 2.0f

// ---------------- K0: zero output + denom ----------------
__global__ void zero_kernel(float* __restrict__ out, float* __restrict__ denom) {
    int tid = blockIdx.x * blockDim.x + threadIdx.x;
    int stride = gridDim.x * blockDim.x;
    for (int i = tid; i < V_NUM * DIM; i += stride) out[i] = 0.0f;
    for (int i = tid; i < V_NUM; i += stride) denom[i] = 0.0f;
}

// ---------------- K1: fused GEMVs via V_WMMA_F32_16X16X4_F32 ----------------
// One wave handles 16 rows. svt[i] = v_fea[i]·a_v + t_fea[i]·a_t  (single WMMA
// accumulator chain over 2*32 k-steps); se[j] = e_fea[j]·a_e.
// A (16x4 f32): lane L<16 holds row L, K={k0,k0+1}; lane L>=16 holds row L-16,
// K={k0+2,k0+3}. B (4x16) is the projection-vector chunk broadcast across all
// 16 columns, so D[m][n] == dot(row m) for every n.
__global__ __launch_bounds__(256) void dot_wmma_kernel(
    const float* __restrict__ v_fea, const float* __restrict__ t_fea,
    const float* __restrict__ e_fea, const float* __restrict__ a_v,
    const float* __restrict__ a_t,  const float* __restrict__ a_e,
    float* __restrict__ svt, float* __restrict__ se)
{
    const int lane  = threadIdx.x & 31;
    const int wid   = (blockIdx.x * blockDim.x + threadIdx.x) >> 5; // wave id (uniform per wave)
    const int mrow  = lane & 15;            // row within 16-row tile
    const int koff  = (lane >> 4) << 1;     // 0 for lanes 0-15, 2 for lanes 16-31
    const int vWaves = V_NUM / 16;          // 6250 (exact)

    v8f c = {};

    if (wid < vWaves) {
        const int base = wid * 16;
        const float* rowp = v_fea + (size_t)(base + mrow) * DIM;
        #pragma unroll 4
        for (int k0 = 0; k0 < DIM; k0 += 4) {
            v2f a = *(const v2f*)(rowp + k0 + koff);
            v2f b = *(const v2f*)(a_v  + k0 + koff);
            c = __builtin_amdgcn_wmma_f32_16x16x4_f32(false, a, false, b,
                                                      (short)0, c, false, false);
        }
        rowp = t_fea + (size_t)(base + mrow) * DIM;
        #pragma unroll 4
        for (int k0 = 0; k0 < DIM; k0 += 4) {
            v2f a = *(const v2f*)(rowp + k0 + koff);
            v2f b = *(const v2f*)(a_t  + k0 + koff);
            c = __builtin_amdgcn_wmma_f32_16x16x4_f32(false, a, false, b,
                                                      (short)0, c, false, false);
        }
        // D layout: VGPR i = rows {i, 8+i}; lanes 0-15 -> rows 0..7, 16-31 -> 8..15.
        if (mrow == 0) {
            float* o = svt + base + (lane >> 4) * 8;
            #pragma unroll
            for (int i = 0; i < 8; ++i) o[i] = c[i];
        }
    } else {
        const int base = (wid - vWaves) * 16;
        if (base < E_NUM) {                  // E_NUM/16 = 3125 waves (exact)
            const float* rowp = e_fea + (size_t)(base + mrow) * DIM;
            #pragma unroll 4
            for (int k0 = 0; k0 < DIM; k0 += 4) {
                v2f a = *(const v2f*)(rowp + k0 + koff);
                v2f b = *(const v2f*)(a_e  + k0 + koff);
                c = __builtin_amdgcn_wmma_f32_16x16x4_f32(false, a, false, b,
                                                          (short)0, c, false, false);
            }
            if (mrow == 0) {
                float* o = se + base + (lane >> 4) * 8;
                #pragma unroll
                for (int i = 0; i < 8; ++i) o[i] = c[i];
            }
        }
    }
}

// ---------------- K2: per-pair exp + denom accumulation ----------------
// tanh bounds logits to [-2,2] => exp(a) in [0.135, 7.39]; the reference's
// segment_max is pure overflow protection and can be dropped safely.
__global__ void pair_softmax_kernel(const int* __restrict__ ve,
                                    const float* __restrict__ svt,
                                    const float* __restrict__ se,
                                    float* __restrict__ exbuf,
                                    float* __restrict__ denom)
{
    int p = blockIdx.x * blockDim.x + threadIdx.x;
    if (p >= N_PAIRS) return;
    int v = ve[2 * p];
    int e = ve[2 * p + 1];
    float a  = tanhf(svt[v] + se[e]) * INV_TAU;
    float ex = __expf(a);
    exbuf[p] = ex;
    unsafeAtomicAdd(&denom[v], ex);        // native global_atomic_add_f32
}

// ---------------- K3: weighted scatter-add of e_fea rows ----------------
// One wave per pair: 32 lanes x float4 = one 512B e_fea row, fully coalesced;
// 4 native f32 atomics per lane into the L2-resident output.
__global__ __launch_bounds__(256) void aggregate_kernel(
    const int* __restrict__ ve, const float* __restrict__ exbuf,
    const float* __restrict__ denom, const float* __restrict__ e_fea,
    float* __restrict__ out)
{
    int tid  = blockIdx.x * blockDim.x + threadIdx.x;
    int p    = tid >> 5;
    if (p >= N_PAIRS) return;
    int lane = tid & 31;
    int v = ve[2 * p];
    int e = ve[2 * p + 1];
    float w = exbuf[p] / denom[v];
    float4 val = ((const float4*)(e_fea + (size_t)e * DIM))[lane];
    float* dst = out + (size_t)v * DIM + lane * 4;
    unsafeAtomicAdd(dst + 0, w * val.x);
    unsafeAtomicAdd(dst + 1, w * val.y);
    unsafeAtomicAdd(dst + 2, w * val.z);
    unsafeAtomicAdd(dst + 3, w * val.w);
}

extern "C" void kernel_launch(void* const* d_in, const int* in_sizes, int n_in,
                              void* d_out, int out_size, void* d_ws, size_t ws_size,
                              hipStream_t stream) {
    const float* v_fea = (const float*)d_in[0];
    const float* t_fea = (const float*)d_in[1];
    const float* e_fea = (const float*)d_in[2];
    const float* a_v   = (const float*)d_in[3];
    const float* a_t   = (const float*)d_in[4];
    const float* a_e   = (const float*)d_in[5];
    const int*   ve    = (const int*)d_in[6];
    float* out = (float*)d_out;

    // workspace layout (floats): svt[V] | se[E] | denom[V] | ex[N_PAIRS]  = 3.4 MB
    float* svt   = (float*)d_ws;
    float* se    = svt + V_NUM;
    float* denom = se + E_NUM;
    float* exbuf = denom + V_NUM;

    zero_kernel<<<2048, 256, 0, stream>>>(out, denom);

    const int waves  = V_NUM / 16 + E_NUM / 16;              // 9375
    const int blocks = (waves * 32 + 255) / 256;             // 1172
    dot_wmma_kernel<<<blocks, 256, 0, stream>>>(v_fea, t_fea, e_fea,
                                                a_v, a_t, a_e, svt, se);

    pair_softmax_kernel<<<(N_PAIRS + 255) / 256, 256, 0, stream>>>(ve, svt, se,
                                                                   exbuf, denom);

    aggregate_kernel<<<(N_PAIRS * 32) / 256, 256, 0, stream>>>(ve, exbuf, denom,
                                                               e_fea, out);
}